// RWKV_RNN_2405181685881
// MI455X (gfx1250) — compile-verified
//
#include <hip/hip_runtime.h>
#include <cmath>

// RWKV-v4 single-token inference for gfx1250 (MI455X).
// HBM-bound (~835 MB f32 weights/token -> ~36 us roofline @ 23.3 TB/s).
// GEMVs on the matrix pipe via V_WMMA_F32_16X16X4_F32:
//   - one 16-row tile per 512-thread block; 16 waves split K (ncols/16 each),
//   - B operand = x-chunk broadcast across all 16 columns (LDS ds_load_b64),
//   - wave partials reduced in LDS; epilogue (sigmoid/relu^2/res/gate) fused,
//   - unroll 16 -> 16 outstanding global_load_b64 per wave per clause.
// Elementwise stages (LN+token-shift mix, WKV) recomputed as per-block
// prologues, fusing each layer into 4 kernel launches (50 launches total).

#define RWKV_V 50277
#define RWKV_D 1024
#define RWKV_L 12
#define RWKV_H 4096

#define BS  512              // threads per block
#define NW  16               // waves per block (wave32)

typedef __attribute__((ext_vector_type(2))) float v2f;
typedef __attribute__((ext_vector_type(8))) float v8f;

// ---------------------------------------------------------------------------
// Block-wide mean/rstd over 1024 elements (2 per thread, 512 threads).
// ---------------------------------------------------------------------------
__device__ __forceinline__ float2 block_mean_rstd(float2 v, float* red1, float* red2) {
    float s1 = v.x + v.y;
    float s2 = v.x * v.x + v.y * v.y;
    for (int off = 16; off > 0; off >>= 1) {
        s1 += __shfl_xor(s1, off, 32);
        s2 += __shfl_xor(s2, off, 32);
    }
    const int lane = threadIdx.x & 31, wid = threadIdx.x >> 5;
    if (lane == 0) { red1[wid] = s1; red2[wid] = s2; }
    __syncthreads();
    float t1 = 0.f, t2 = 0.f;
#pragma unroll
    for (int i = 0; i < NW; ++i) { t1 += red1[i]; t2 += red2[i]; }
    const float mu  = t1 * (1.0f / RWKV_D);
    const float var = t2 * (1.0f / RWKV_D) - mu * mu;
    return make_float2(mu, rsqrtf(var + 1e-5f));
}

// ---------------------------------------------------------------------------
// One 16-row GEMV tile per block via WMMA f32 16x16x4, 16-way split-K.
// xlds: the ncols-long input vector staged in LDS.
// mode: 0 plain, 1 sigmoid, 2 relu^2, 3 res+, 4 res + gate*.
// ---------------------------------------------------------------------------
__device__ __forceinline__ void gemv_tile(const float* __restrict__ W,
                                          const float* __restrict__ xlds,
                                          float* __restrict__ y,
                                          int nrows, int ncols, int mode,
                                          const float* __restrict__ res,
                                          const float* __restrict__ gate,
                                          float (*part)[16]) {
    const int tid  = threadIdx.x;
    const int lane = tid & 31, wv = tid >> 5;
    const int half = lane >> 4;    // lane group: K={0,1} vs K={2,3} per wmma step
    const int mrow = lane & 15;    // A-matrix row within tile
    const int rb   = blockIdx.x;
    const int nrb  = (nrows + 15) >> 4;
    const int chunk = ncols >> 4;  // per-wave K extent (multiple of 4)

    if (rb < nrb) {                                  // block-uniform
        int r = rb * 16 + mrow;
        r = (r < nrows) ? r : (nrows - 1);           // branchless clamp, EXEC full
        const float* wr = W + (size_t)r * ncols + wv * chunk + 2 * half;
        const float* bx = xlds + wv * chunk + 2 * half;
        v8f acc = {0.f, 0.f, 0.f, 0.f, 0.f, 0.f, 0.f, 0.f};
#pragma unroll 16
        for (int kk = 0; kk < chunk; kk += 4) {
            const v2f a = *reinterpret_cast<const v2f*>(wr + kk);  // global_load_b64
            const v2f b = *reinterpret_cast<const v2f*>(bx + kk);  // ds_load_b64 broadcast
            acc = __builtin_amdgcn_wmma_f32_16x16x4_f32(
                false, a, false, b, (short)0, acc, false, false);
        }
        // All 16 D-columns identical; lanes 0 / 16 carry rows 0..7 / 8..15.
        if (mrow == 0) {
#pragma unroll
            for (int j = 0; j < 8; ++j) part[wv][half * 8 + j] = acc[j];
        }
    }
    __syncthreads();
    if (rb < nrb && tid < 16) {
        const int row = rb * 16 + tid;
        if (row < nrows) {
            float s = 0.f;
#pragma unroll
            for (int w16 = 0; w16 < NW; ++w16) s += part[w16][tid];  // deterministic
            if (mode == 1)      s = 1.f / (1.f + expf(-s));
            else if (mode == 2) { const float t = fmaxf(s, 0.f); s = t * t; }
            else if (mode == 3) s = res[row] + s;
            else if (mode == 4) s = res[row] + gate[row] * s;
            y[row] = s;
        }
    }
}

// ---------------------------------------------------------------------------
// x = LN(emb[token])   (single block)
// ---------------------------------------------------------------------------
__global__ void rwkv_ln0(const int* __restrict__ token, const float* __restrict__ emb,
                         const float* __restrict__ w, const float* __restrict__ b,
                         float* __restrict__ xout) {
    __shared__ float red1[NW], red2[NW];
    const int tid = threadIdx.x;
    const float* src = emb + (size_t)token[0] * RWKV_D;
    float2 v = reinterpret_cast<const float2*>(src)[tid];
    const float2 ms = block_mean_rstd(v, red1, red2);
    const int base = tid * 2;
    float a2[2] = {v.x, v.y};
#pragma unroll
    for (int j = 0; j < 2; ++j) {
        const int i = base + j;
        xout[i] = (a2[j] - ms.x) * ms.y * w[i] + b[i];
    }
}

// ---------------------------------------------------------------------------
// Layer stage 1: LN1 + token-shift mix prologue, then k/v/r GEMVs.
// grid = (64, 3): y selects (mk,kw)->k, (mv,vw)->v, (mr,rw)->sigmoid r.
// Block (0,0) also writes xn into new-state slot 1.
// ---------------------------------------------------------------------------
__global__ void rwkv_att_qkv(const float* __restrict__ x,
                             const float* __restrict__ lw, const float* __restrict__ lb,
                             const float* __restrict__ prev,
                             const float* __restrict__ mk, const float* __restrict__ mv,
                             const float* __restrict__ mr,
                             const float* __restrict__ kw, const float* __restrict__ vw,
                             const float* __restrict__ rw,
                             float* __restrict__ kout, float* __restrict__ vout,
                             float* __restrict__ rout,
                             float* __restrict__ xn_state) {
    __shared__ float red1[NW], red2[NW];
    __shared__ float part[NW][16];
    extern __shared__ float xmix[];
    const int tid = threadIdx.x;
    float2 v = reinterpret_cast<const float2*>(x)[tid];
    const float2 ms = block_mean_rstd(v, red1, red2);
    const int sel = blockIdx.y;
    const float* mm = (sel == 0) ? mk : (sel == 1) ? mv : mr;
    const float* W  = (sel == 0) ? kw : (sel == 1) ? vw : rw;
    float* y        = (sel == 0) ? kout : (sel == 1) ? vout : rout;
    const int mode  = (sel == 2) ? 1 : 0;
    const int base = tid * 2;
    float a2[2] = {v.x, v.y};
#pragma unroll
    for (int j = 0; j < 2; ++j) {
        const int i = base + j;
        const float xn = (a2[j] - ms.x) * ms.y * lw[i] + lb[i];
        if (sel == 0 && blockIdx.x == 0) xn_state[i] = xn;
        const float m = mm[i];
        xmix[i] = xn * m + prev[i] * (1.f - m);
    }
    __syncthreads();
    gemv_tile(W, xmix, y, RWKV_D, RWKV_D, mode, nullptr, nullptr, part);
}

// ---------------------------------------------------------------------------
// Layer stage 2: WKV recurrence prologue (block 0 writes aa2/bb2/p2 state),
// then sx = x + ow @ wkv.   grid = (64)
// ---------------------------------------------------------------------------
__global__ void rwkv_att_out(const float* __restrict__ kbuf, const float* __restrict__ vbuf,
                             const float* __restrict__ rbuf,
                             const float* __restrict__ tf, const float* __restrict__ tdraw,
                             const float* __restrict__ aa, const float* __restrict__ bb,
                             const float* __restrict__ pp,
                             float* __restrict__ aa2, float* __restrict__ bb2,
                             float* __restrict__ pp2,
                             const float* __restrict__ ow, const float* __restrict__ xres,
                             float* __restrict__ sx) {
    __shared__ float part[NW][16];
    extern __shared__ float wkv[];
    const int tid = threadIdx.x;
#pragma unroll
    for (int j = 0; j < 2; ++j) {
        const int i = tid * 2 + j;
        const float kk = kbuf[i], vv = vbuf[i];
        const float aao = aa[i], bbo = bb[i], ppo = pp[i];
        const float ww = tf[i] + kk;
        const float pm = fmaxf(ppo, ww);
        const float e1 = expf(ppo - pm), e2 = expf(ww - pm);
        wkv[i] = rbuf[i] * (e1 * aao + e2 * vv) / (e1 * bbo + e2);
        if (blockIdx.x == 0) {
            const float td  = -expf(tdraw[i]);
            const float ww2 = ppo + td;
            const float p2  = fmaxf(ww2, kk);
            const float f1 = expf(ww2 - p2), f2 = expf(kk - p2);
            aa2[i] = f1 * aao + f2 * vv;
            bb2[i] = f1 * bbo + f2;
            pp2[i] = p2;
        }
    }
    __syncthreads();
    gemv_tile(ow, wkv, sx, RWKV_D, RWKV_D, 3, xres, nullptr, part);
}

// ---------------------------------------------------------------------------
// Layer stage 3: LN2 + mix prologue, then frw->sigmoid r2 (y=0) or
// fkw->relu^2 kq (y=1).   grid = (256, 2); y=0 blocks with x>=64 idle.
// Block (0,0) writes xn2 into new-state slot 0.
// ---------------------------------------------------------------------------
__global__ void rwkv_ffn_rk(const float* __restrict__ sx,
                            const float* __restrict__ lw, const float* __restrict__ lb,
                            const float* __restrict__ prev0,
                            const float* __restrict__ fmk, const float* __restrict__ fmr,
                            const float* __restrict__ frw, const float* __restrict__ fkw,
                            float* __restrict__ r2out, float* __restrict__ kqout,
                            float* __restrict__ xn2_state) {
    __shared__ float red1[NW], red2[NW];
    __shared__ float part[NW][16];
    extern __shared__ float xmix[];
    const int tid = threadIdx.x;
    float2 v = reinterpret_cast<const float2*>(sx)[tid];
    const float2 ms = block_mean_rstd(v, red1, red2);
    const int sel = blockIdx.y;              // 0 -> r path, 1 -> k path
    const float* mm = sel ? fmk : fmr;
    const int base = tid * 2;
    float a2[2] = {v.x, v.y};
#pragma unroll
    for (int j = 0; j < 2; ++j) {
        const int i = base + j;
        const float xn = (a2[j] - ms.x) * ms.y * lw[i] + lb[i];
        if (sel == 0 && blockIdx.x == 0) xn2_state[i] = xn;
        const float m = mm[i];
        xmix[i] = xn * m + prev0[i] * (1.f - m);
    }
    __syncthreads();
    if (sel == 0) gemv_tile(frw, xmix, r2out, RWKV_D, RWKV_D, 1, nullptr, nullptr, part);
    else          gemv_tile(fkw, xmix, kqout, RWKV_H, RWKV_D, 2, nullptr, nullptr, part);
}

// ---------------------------------------------------------------------------
// Layer stage 4: x = sx + r2 * (fvw @ kq).   grid = (64)
// ---------------------------------------------------------------------------
__global__ void rwkv_ffn_out(const float* __restrict__ fvw, const float* __restrict__ kq,
                             const float* __restrict__ sx, const float* __restrict__ r2,
                             float* __restrict__ xout) {
    __shared__ float part[NW][16];
    extern __shared__ float xlds[];
    for (int i = threadIdx.x; i < RWKV_H; i += BS) xlds[i] = kq[i];
    __syncthreads();
    gemv_tile(fvw, xlds, xout, RWKV_D, RWKV_H, 4, sx, r2, part);
}

// ---------------------------------------------------------------------------
// Head: LN_out prologue, then logits = head @ ln(x).   grid = (3143)
// ---------------------------------------------------------------------------
__global__ void rwkv_head(const float* __restrict__ x, const float* __restrict__ lw,
                          const float* __restrict__ lb, const float* __restrict__ head,
                          float* __restrict__ logits) {
    __shared__ float red1[NW], red2[NW];
    __shared__ float part[NW][16];
    extern __shared__ float xlds[];
    const int tid = threadIdx.x;
    float2 v = reinterpret_cast<const float2*>(x)[tid];
    const float2 ms = block_mean_rstd(v, red1, red2);
    const int base = tid * 2;
    float a2[2] = {v.x, v.y};
#pragma unroll
    for (int j = 0; j < 2; ++j) {
        const int i = base + j;
        xlds[i] = (a2[j] - ms.x) * ms.y * lw[i] + lb[i];
    }
    __syncthreads();
    gemv_tile(head, xlds, logits, RWKV_V, RWKV_D, 0, nullptr, nullptr, part);
}

// ---------------------------------------------------------------------------
extern "C" void kernel_launch(void* const* d_in, const int* in_sizes, int n_in,
                              void* d_out, int out_size, void* d_ws, size_t ws_size,
                              hipStream_t stream) {
    (void)in_sizes; (void)n_in; (void)out_size; (void)ws_size;
    const int D = RWKV_D, L = RWKV_L, V = RWKV_V;
    const size_t DD = (size_t)D * D, HD = (size_t)RWKV_H * D;

    const int*   token = (const int*)d_in[0];
    const float* state = (const float*)d_in[1];
    const float* emb   = (const float*)d_in[2];
    const float* ln0w  = (const float*)d_in[3];
    const float* ln0b  = (const float*)d_in[4];
    const float* ln1w  = (const float*)d_in[5];
    const float* ln1b  = (const float*)d_in[6];
    const float* ln2w  = (const float*)d_in[7];
    const float* ln2b  = (const float*)d_in[8];
    const float* tdr   = (const float*)d_in[9];   // raw time_decay
    const float* tf    = (const float*)d_in[10];  // time_first
    const float* amk   = (const float*)d_in[11];
    const float* amv   = (const float*)d_in[12];
    const float* amr   = (const float*)d_in[13];
    const float* kw    = (const float*)d_in[14];
    const float* vw    = (const float*)d_in[15];
    const float* rw    = (const float*)d_in[16];
    const float* ow    = (const float*)d_in[17];
    const float* fmk   = (const float*)d_in[18];
    const float* fmr   = (const float*)d_in[19];
    const float* fkw   = (const float*)d_in[20];
    const float* frw   = (const float*)d_in[21];
    const float* fvw   = (const float*)d_in[22];
    const float* lnow  = (const float*)d_in[23];
    const float* lnob  = (const float*)d_in[24];
    const float* head  = (const float*)d_in[25];

    float* logits = (float*)d_out;   // [V]
    float* nstate = logits + V;      // [5*L*D], layer i rows: xn2, xn, aa2, bb2, p2

    float* ws   = (float*)d_ws;
    float* x    = ws + 0 * D;        // hidden (updated per layer)
    float* kbuf = ws + 1 * D;
    float* vbuf = ws + 2 * D;
    float* rbuf = ws + 3 * D;
    float* sx   = ws + 4 * D;
    float* r2   = ws + 5 * D;
    float* kq   = ws + 6 * D;        // [H]

    const size_t shmD = (size_t)D * sizeof(float);
    const size_t shmH = (size_t)RWKV_H * sizeof(float);

    rwkv_ln0<<<1, BS, 0, stream>>>(token, emb, ln0w, ln0b, x);

    for (int i = 0; i < L; ++i) {
        const float* st  = state  + (size_t)(5 * i) * D;
        float*       nst = nstate + (size_t)(5 * i) * D;

        rwkv_att_qkv<<<dim3(64, 3), BS, shmD, stream>>>(
            x, ln1w + (size_t)i * D, ln1b + (size_t)i * D, st + D,
            amk + (size_t)i * D, amv + (size_t)i * D, amr + (size_t)i * D,
            kw + (size_t)i * DD, vw + (size_t)i * DD, rw + (size_t)i * DD,
            kbuf, vbuf, rbuf, nst + D);

        rwkv_att_out<<<dim3(64), BS, shmD, stream>>>(
            kbuf, vbuf, rbuf, tf + (size_t)i * D, tdr + (size_t)i * D,
            st + 2 * D, st + 3 * D, st + 4 * D,
            nst + 2 * D, nst + 3 * D, nst + 4 * D,
            ow + (size_t)i * DD, x, sx);

        rwkv_ffn_rk<<<dim3(256, 2), BS, shmD, stream>>>(
            sx, ln2w + (size_t)i * D, ln2b + (size_t)i * D, st,
            fmk + (size_t)i * D, fmr + (size_t)i * D,
            frw + (size_t)i * DD, fkw + (size_t)i * HD,
            r2, kq, nst);

        rwkv_ffn_out<<<dim3(64), BS, shmH, stream>>>(
            fvw + (size_t)i * HD, kq, sx, r2, x);
    }

    rwkv_head<<<dim3((RWKV_V + 15) / 16), BS, shmD, stream>>>(
        x, lnow, lnob, head, logits);
}